// HistByNorm_40673340293939
// MI455X (gfx1250) — compile-verified
//
#include <hip/hip_runtime.h>

// HistByNorm: out[n][j] = coeff_j * mean_d exp(-0.5*((x[n][d]-c_j)/w_j)^2)
// N = 2048 rows, D = 8192, NBINS = 16.
// exp-bound kernel: inner loop = v_fma + v_mul(neg) + v_exp_f32 + v_add per (elem,bin).
// Block reduction done on the matrix pipe via chained V_WMMA_F32_16X16X4_F32.

typedef __attribute__((ext_vector_type(2))) float v2f;
typedef __attribute__((ext_vector_type(8))) float v8f;

#define NBINS 16
#define TPB   256

__device__ __forceinline__ float fast_exp2(float a) {
#if __has_builtin(__builtin_amdgcn_exp2f)
  return __builtin_amdgcn_exp2f(a);   // v_exp_f32
#else
  return exp2f(a);
#endif
}

__device__ __forceinline__ float fast_rcp(float a) {
#if __has_builtin(__builtin_amdgcn_rcpf)
  return __builtin_amdgcn_rcpf(a);    // v_rcp_f32 (1 ulp; widths are benign scales)
#else
  return 1.0f / a;
#endif
}

__global__ __launch_bounds__(TPB) void HistByNorm_40673340293939_kernel(
    const float* __restrict__ x,
    const float* __restrict__ centers,
    const float* __restrict__ widths,
    float* __restrict__ out,
    int D, float invD)
{
  __shared__ float P[TPB * NBINS];   // 16 KB: per-thread partial sums

  const int n = blockIdx.x;
  const int t = threadIdx.x;
  const float* __restrict__ xrow = x + (size_t)n * (size_t)D;

  // z' = x*a_j - b_j  with a_j = s/w_j, b_j = c_j*s/w_j, s = sqrt(0.5*log2(e)).
  // Then exp(-0.5*((x-c)/w)^2) == exp2(-(z'^2)).
  const float S = 0.84936059f; // sqrt(0.72134752)
  float aj[NBINS], bj[NBINS];
#pragma unroll
  for (int j = 0; j < NBINS; ++j) {
    float w = widths[j];       // uniform -> scalar loads
    float c = centers[j];
    float iw = S * fast_rcp(w);
    aj[j] = iw;
    bj[j] = c * iw;
  }

  float acc[NBINS];
#pragma unroll
  for (int j = 0; j < NBINS; ++j) acc[j] = 0.0f;

  // Main streaming loop: coalesced float4 loads, 16 exps per element.
  const float4* __restrict__ xrow4 = (const float4*)xrow;
  const int iters = D / (TPB * 4);
  for (int k = 0; k < iters; ++k) {
    float4 v = xrow4[t + k * TPB];
    if (k + 1 < iters)
      __builtin_prefetch(&xrow4[t + (k + 1) * TPB], 0, 0); // global_prefetch
    float xs[4] = {v.x, v.y, v.z, v.w};
#pragma unroll
    for (int e = 0; e < 4; ++e) {
      const float xv = xs[e];
#pragma unroll
      for (int j = 0; j < NBINS; ++j) {
        float z = __builtin_fmaf(xv, aj[j], -bj[j]);
        acc[j] += fast_exp2(z * (-z));   // neg folds into v_mul src modifier
      }
    }
  }
  // Generic tail (D not multiple of 1024)
  for (int d = iters * TPB * 4 + t; d < D; d += TPB) {
    const float xv = xrow[d];
#pragma unroll
    for (int j = 0; j < NBINS; ++j) {
      float z = __builtin_fmaf(xv, aj[j], -bj[j]);
      acc[j] += fast_exp2(z * (-z));
    }
  }

  // Stage partials: P[t][j]
#pragma unroll
  for (int j = 0; j < NBINS; ++j) P[t * NBINS + j] = acc[j];
  __syncthreads();

  // Wave 0 reduces P (256x16) over threads with chained f32 WMMAs:
  // D[m][n] = sum_k A[m][k] * 1 + C,  A[m][k] = P[r0+k][m].
  // A 16x4 f32 layout: lanes 0-15 hold K=0(vgpr0),K=1(vgpr1); lanes 16-31 K=2,K=3.
  if (t < 32) {
    const int lane = t;
    const int half = lane >> 4;
    const int m    = lane & 15;
    v8f cacc = {};
    v2f ones; ones.x = 1.0f; ones.y = 1.0f;
#pragma unroll 8
    for (int r0 = 0; r0 < TPB; r0 += 4) {
      v2f a;
      a.x = P[(r0 + 2 * half + 0) * NBINS + m];
      a.y = P[(r0 + 2 * half + 1) * NBINS + m];
      cacc = __builtin_amdgcn_wmma_f32_16x16x4_f32(
          /*neg_a=*/false, a, /*neg_b=*/false, ones,
          /*c_mod=*/(short)0, cacc, /*reuse_a=*/false, /*reuse_b=*/false);
    }
    // C/D layout: vgpr r -> bin r (lanes 0-15) / bin r+8 (lanes 16-31); all
    // columns hold the same sum, so lane 0 and lane 16 write 8 bins each.
    if (m == 0) {
      const int binBase = half * 8;
      const float INV_SQRT_2PI = 0.39894228040143268f;
#pragma unroll
      for (int r = 0; r < 8; ++r) {
        const int b = binBase + r;
        const float scale = INV_SQRT_2PI * fast_rcp(widths[b]) * invD;
        out[(size_t)n * NBINS + b] = cacc[r] * scale;
      }
    }
  }
}

extern "C" void kernel_launch(void* const* d_in, const int* in_sizes, int n_in,
                              void* d_out, int out_size, void* d_ws, size_t ws_size,
                              hipStream_t stream) {
  const float* x       = (const float*)d_in[0];
  const float* centers = (const float*)d_in[1];
  const float* widths  = (const float*)d_in[2];
  float*       out     = (float*)d_out;

  const int nbins = in_sizes[1];          // 16
  const int N     = out_size / nbins;     // B*C = 2048
  const int D     = in_sizes[0] / N;      // 8192
  const float invD = 1.0f / (float)D;

  HistByNorm_40673340293939_kernel<<<N, TPB, 0, stream>>>(
      x, centers, widths, out, D, invD);
}